// FSNeuron_80058190397914
// MI455X (gfx1250) — compile-verified
//
#include <hip/hip_runtime.h>
#include <stdint.h>

// FS-neuron streaming kernel for MI455X (gfx1250).
// Roofline: read 512 MiB + write 512 MiB ~= 1.07 GB -> ~46us floor at 23.3 TB/s.
// No matrix structure -> no WMMA; the CDNA5 path used is the async global->LDS
// data mover (GLOBAL_LOAD_ASYNC_TO_LDS_B128, ASYNCcnt) with per-wave double
// buffering and split waits (no workgroup barriers: each lane reads back only
// its own staged LDS slots).

typedef float v4f __attribute__((ext_vector_type(4)));

#define TSTEPS 8
#define BLOCK  256
#define TSTRIDE 4096u          // bytes per timestep slab in LDS (256 lanes * 16B)
#define PSTRIDE 32768u         // bytes per parity buffer (8 slabs)

static __device__ __forceinline__ uint32_t lds_off(const void* p) {
    // Flat LDS addresses are {SHARED_BASE[63:32], lds_offset[31:0]} on CDNA5:
    // truncation yields the LDS byte offset.
    return (uint32_t)(uintptr_t)p;
}

static __device__ __forceinline__ void async_ld_b128_gvs(uint32_t dst_lds,
                                                         uint32_t voff,
                                                         const v4f* sbase) {
    // GVS mode: SGPR-pair base + signed 32-bit VGPR byte offset.
    // Per-lane 16B global -> per-lane LDS slot; tracked by ASYNCcnt.
    asm volatile("global_load_async_to_lds_b128 %0, %1, %2"
                 :
                 : "v"(dst_lds), "v"(voff), "s"(sbase)
                 : "memory");
}

__global__ __launch_bounds__(BLOCK)
void fs_neuron_main(const float* __restrict__ x,
                    const float* __restrict__ dg,
                    const float* __restrict__ hg,
                    const float* __restrict__ tg,
                    float* __restrict__ y,
                    long long n4,        // per-timestep element count / 4
                    int numTiles)
{
    __shared__ v4f smem[2][TSTEPS][BLOCK];   // 64 KiB double-buffered tile

    const int tid = threadIdx.x;
    v4f* __restrict__ y4 = (v4f*)y;

    // Uniform per-timestep global bases -> SGPR pairs for the GVS asm.
    const v4f* xb[TSTEPS];
#pragma unroll
    for (int t = 0; t < TSTEPS; ++t)
        xb[t] = (const v4f*)x + (long long)t * n4;

    // Per-step scales d^-(t+1), h^-(t+1), theta^-(t+1) via cumulative 1/v products.
    const float rd = 1.0f / dg[0];
    const float rh = 1.0f / hg[0];
    const float rt = 1.0f / tg[0];

    int tile = blockIdx.x;
    if (tile >= numTiles) return;

    // Single LDS base; slot(par, t) = lds0 + par*PSTRIDE + t*TSTRIDE.
    const uint32_t lds0 = lds_off(&smem[0][0][tid]);

    // Loop-carried addressing state: one 32-bit byte offset for loads (plane is
    // 64 MiB < 2^31 so the signed GVS offset is safe) and a 64-bit element
    // index for stores.
    uint32_t voff = (uint32_t)(((long long)tile * BLOCK + tid) * sizeof(v4f));
    const uint32_t vstep = (uint32_t)gridDim.x * BLOCK * (uint32_t)sizeof(v4f);
    long long e = (long long)tile * BLOCK + tid;
    const long long estep = (long long)gridDim.x * BLOCK;

    // Prologue: stage first tile into buffer 0 (8 async B128 loads / lane).
#pragma unroll
    for (int t = 0; t < TSTEPS; ++t)
        async_ld_b128_gvs(lds0 + (uint32_t)t * TSTRIDE, voff, xb[t]);

    int par = 0;
    for (; tile < numTiles; tile += (int)gridDim.x) {
        if (tile + (int)gridDim.x < numTiles) {
            // Issue next tile into the other buffer, then wait until only those
            // 8 remain outstanding (async loads complete in order) -> current
            // buffer fully resident in LDS.
            voff += vstep;
            const uint32_t nb = lds0 + (uint32_t)(par ^ 1) * PSTRIDE;
#pragma unroll
            for (int t = 0; t < TSTEPS; ++t)
                async_ld_b128_gvs(nb + (uint32_t)t * TSTRIDE, voff, xb[t]);
            asm volatile("s_wait_asynccnt 0x8" ::: "memory");
        } else {
            asm volatile("s_wait_asynccnt 0x0" ::: "memory");
        }

        // v = sum over T from this lane's own staged slots.
        v4f v = smem[par][0][tid];
#pragma unroll
        for (int t = 1; t < TSTEPS; ++t)
            v += smem[par][t][tid];

        v4f sg;
#pragma unroll
        for (int c = 0; c < 4; ++c) {
            float a = v[c];
            sg[c] = (a > 0.0f) ? 1.0f : ((a < 0.0f) ? -1.0f : 0.0f);
            v[c]  = __builtin_fabsf(a);
        }

        // Sequential threshold-subtract recurrence; stream spikes out with NT
        // stores (1 GB stream >> 192 MB L2, nothing re-read).
        float dc = 1.0f, hc = 1.0f, tc = 1.0f;
#pragma unroll
        for (int t = 0; t < TSTEPS; ++t) {
            dc *= rd; hc *= rh; tc *= rt;
            v4f o;
#pragma unroll
            for (int c = 0; c < 4; ++c) {
                float s = (v[c] > tc) ? 1.0f : 0.0f;
                v[c] -= hc * s;
                o[c]  = sg[c] * dc * s;
            }
            __builtin_nontemporal_store(o, y4 + (long long)t * n4 + e);
        }

        e += estep;
        par ^= 1;
    }
}

// Scalar tail for element counts not divisible by 1024 (not hit for the
// reference shape; kept for generality).
__global__ void fs_neuron_tail(const float* __restrict__ x,
                               const float* __restrict__ dg,
                               const float* __restrict__ hg,
                               const float* __restrict__ tg,
                               float* __restrict__ y,
                               long long N, long long base, long long rem)
{
    long long g = (long long)blockIdx.x * blockDim.x + threadIdx.x;
    if (g >= rem) return;
    long long i = base + g;

    float v = 0.0f;
#pragma unroll
    for (int t = 0; t < TSTEPS; ++t) v += x[(long long)t * N + i];

    float sg = (v > 0.0f) ? 1.0f : ((v < 0.0f) ? -1.0f : 0.0f);
    v = __builtin_fabsf(v);

    const float rd = 1.0f / dg[0];
    const float rh = 1.0f / hg[0];
    const float rt = 1.0f / tg[0];
    float dc = 1.0f, hc = 1.0f, tc = 1.0f;
#pragma unroll
    for (int t = 0; t < TSTEPS; ++t) {
        dc *= rd; hc *= rh; tc *= rt;
        float s = (v > tc) ? 1.0f : 0.0f;
        v -= hc * s;
        y[(long long)t * N + i] = sg * dc * s;
    }
}

extern "C" void kernel_launch(void* const* d_in, const int* in_sizes, int n_in,
                              void* d_out, int out_size, void* d_ws, size_t ws_size,
                              hipStream_t stream) {
    const float* x  = (const float*)d_in[0];   // [T, 2, 2048, 4096] fp32
    const float* dg = (const float*)d_in[1];   // [1]
    const float* hg = (const float*)d_in[2];   // [1]
    const float* tg = (const float*)d_in[3];   // [1]
    float* y = (float*)d_out;                  // [T, 2, 2048, 4096] fp32

    const long long total = (long long)in_sizes[0];   // T * N
    const long long N  = total / TSTEPS;              // per-timestep elements
    const long long n4 = N / 4;                       // float4 units
    const int numTiles = (int)(n4 / BLOCK);           // 1024-float tiles

    if (numTiles > 0) {
        int grid = numTiles < 4096 ? numTiles : 4096; // ~4 tiles/block -> pipeline depth
        fs_neuron_main<<<grid, BLOCK, 0, stream>>>(x, dg, hg, tg, y, n4, numTiles);
    }

    const long long done = (long long)numTiles * BLOCK * 4;
    const long long rem  = N - done;
    if (rem > 0) {
        int tgrid = (int)((rem + 255) / 256);
        fs_neuron_tail<<<tgrid, 256, 0, stream>>>(x, dg, hg, tg, y, N, done, rem);
    }
}